// EKF_17532056502322
// MI455X (gfx1250) — compile-verified
//
#include <hip/hip_runtime.h>
#include <stdint.h>

// EKF over T=32768 timesteps. Serial scan (latency-bound): one wave runs the
// chain; inputs are staged global->LDS with CDNA5 async-to-LDS DMA, double
// buffered via s_wait_asynccnt. Covariance recursion reduced algebraically to
// a scalar (Qn, Rn, p0 are all scalar*I => p stays scalar*I exactly).

namespace {
constexpr int   kT  = 32768;
constexpr float kDT = 0.01f;
constexpr float kPI = 3.14159f;
constexpr float kQn = 1e-10f;
constexpr float kRn = 0.0005f;

__device__ __forceinline__ void qmul(float aw, float ax, float ay, float az,
                                     float bw, float bx, float by, float bz,
                                     float& rw, float& rx, float& ry, float& rz) {
  rw = aw*bw - ax*bx - ay*by - az*bz;
  rx = aw*bx + ax*bw + ay*bz - az*by;
  ry = aw*by - ax*bz + ay*bw + az*bx;
  rz = aw*bz + ax*by - ay*bx + az*bw;
}

// Issue 7 async global->LDS dword copies for one step per lane:
// omega (3 consecutive dwords) and z (4 consecutive dwords).
// ISA: INST_OFFSET is added to BOTH the LDS and the global address, which
// matches our consecutive-dword layouts on both sides.
__device__ __forceinline__ void async_issue(uint32_t ldsW, uint32_t ldsZ,
                                            const float* gx, const float* gz) {
  asm volatile(
      "global_load_async_to_lds_b32 %0, %2, off\n\t"
      "global_load_async_to_lds_b32 %0, %2, off offset:4\n\t"
      "global_load_async_to_lds_b32 %0, %2, off offset:8\n\t"
      "global_load_async_to_lds_b32 %1, %3, off\n\t"
      "global_load_async_to_lds_b32 %1, %3, off offset:4\n\t"
      "global_load_async_to_lds_b32 %1, %3, off offset:8\n\t"
      "global_load_async_to_lds_b32 %1, %3, off offset:12"
      :
      : "v"(ldsW), "v"(ldsZ), "v"(gx), "v"(gz)
      : "memory");
}
} // namespace

// Parallel fill of the 6 identically-zero off-diagonal covariance entries
// per output row (d_out is poisoned by the harness; every element must be
// written; diagonals + quaternion are written by the scan kernel).
__global__ void ekf_fill_offdiag(float* __restrict__ out) {
  const int total = kT * 9;
  for (int idx = blockIdx.x * blockDim.x + threadIdx.x; idx < total;
       idx += gridDim.x * blockDim.x) {
    const int t = idx / 9;
    const int j = idx - t * 9;
    if (j != 0 && j != 4 && j != 8) out[(size_t)t * 13 + 4 + j] = 0.0f;
  }
}

__global__ __launch_bounds__(32)
void ekf_scan(const float* __restrict__ x, const float* __restrict__ pred,
              float* __restrict__ out) {
  // [double buffer][step-in-chunk][slot]; slots 0..2 = omega, 3..6 = z, 7 pad
  __shared__ float lds[2][32][8];
  const int lane = threadIdx.x;          // wave32
  const int S    = kT - 1;               // 32767 scan steps
  const int NCH  = (S + 31) >> 5;        // chunks of 32 steps

  // ---- issue chunk 0 (addresses clamped so all 7 ops always issue) ----
  {
    int st = lane; if (st > S - 1) st = S - 1;
    async_issue((uint32_t)(uintptr_t)&lds[0][lane][0],
                (uint32_t)(uintptr_t)&lds[0][lane][3],
                x    + (size_t)st * 6 + 3,
                pred + (size_t)(st + 1) * 7);
  }

  // ---- initial state: xh0 = pred[0,0,:4] (raw), p0 = 1e-10*I ----
  float q0 = pred[0], q1 = pred[1], q2 = pred[2], q3 = pred[3];
  float s = kQn;
  if (lane == 0) {
    out[0] = q0; out[1] = q1; out[2] = q2; out[3] = q3;
    out[4] = kQn; out[8] = kQn; out[12] = kQn;
  }

  for (int c = 0; c < NCH; ++c) {
    const int buf = c & 1;
    if (c + 1 < NCH) {
      // prefetch next chunk, then wait only for the current one (in-order
      // completion of async loads => asynccnt<=7 means chunk c is in LDS)
      int st = (c + 1) * 32 + lane; if (st > S - 1) st = S - 1;
      async_issue((uint32_t)(uintptr_t)&lds[buf ^ 1][lane][0],
                  (uint32_t)(uintptr_t)&lds[buf ^ 1][lane][3],
                  x    + (size_t)st * 6 + 3,
                  pred + (size_t)(st + 1) * 7);
      asm volatile("s_wait_asynccnt 0x7" ::: "memory");
    } else {
      asm volatile("s_wait_asynccnt 0x0" ::: "memory");
    }

    int cnt = S - c * 32; if (cnt > 32) cnt = 32;
    for (int i = 0; i < cnt; ++i) {
      const float w0 = lds[buf][i][0], w1 = lds[buf][i][1], w2 = lds[buf][i][2];
      const float z0 = lds[buf][i][3], z1 = lds[buf][i][4];
      const float z2 = lds[buf][i][5], z3 = lds[buf][i][6];

      // ---- predict: xh += (DT/2)*Sq(xh) @ omega ; normalize ----
      const float h = kDT * 0.5f;
      float n0 = q0 + h * (-q1 * w0 - q2 * w1 - q3 * w2);
      float n1 = q1 + h * ( q0 * w0 - q3 * w1 + q2 * w2);
      float n2 = q2 + h * ( q3 * w0 + q0 * w1 - q1 * w2);
      float n3 = q3 + h * (-q2 * w0 + q1 * w1 + q0 * w2);
      float inm = 1.0f / sqrtf(n0 * n0 + n1 * n1 + n2 * n2 + n3 * n3);
      q0 = n0 * inm; q1 = n1 * inm; q2 = n2 * inm; q3 = n3 * inm;
      s = s + kQn;                       // p = p + Qn   (scalar*I)

      // ---- y = boxminus(z, xh) = 2*log(conj(xh)*z) ----
      float iw, ix, iy, iz;
      qmul(q0, -q1, -q2, -q3, z0, z1, z2, z3, iw, ix, iy, iz);
      const float nv = sqrtf(ix * ix + iy * iy + iz * iz);
      float y0 = 0.0f, y1 = 0.0f, y2 = 0.0f;
      if (nv != 0.0f) {
        float f = (iw == 0.0f) ? (kPI / (2.0f * nv)) : (atanf(nv / iw) / nv);
        f *= 2.0f;
        y0 = f * ix; y1 = f * iy; y2 = f * iz;
      }

      // ---- K = p @ inv(p + Rn) = k*I ----
      const float k = s / (s + kRn);

      // ---- boxplus(xh, k*y) ----
      const float c0 = 0.5f * k * y0, c1 = 0.5f * k * y1, c2 = 0.5f * k * y2;
      const float dn = sqrtf(c0 * c0 + c1 * c1 + c2 * c2);
      const float sinc = (dn == 0.0f) ? 1.0f : (sinf(dn) / dn);
      float e0 = coshf(dn), e1 = sinc * c0, e2 = sinc * c1, e3 = sinc * c2;
      const float ien = 1.0f / sqrtf(e0 * e0 + e1 * e1 + e2 * e2 + e3 * e3);
      e0 *= ien; e1 *= ien; e2 *= ien; e3 *= ien;
      float p0, p1, p2, p3;
      qmul(q0, q1, q2, q3, e0, e1, e2, e3, p0, p1, p2, p3);
      const float ipn = 1.0f / sqrtf(p0 * p0 + p1 * p1 + p2 * p2 + p3 * p3);
      q0 = p0 * ipn; q1 = p1 * ipn; q2 = p2 * ipn; q3 = p3 * ipn;
      const float ixn = 1.0f / sqrtf(q0 * q0 + q1 * q1 + q2 * q2 + q3 * q3);
      q0 *= ixn; q1 *= ixn; q2 *= ixn; q3 *= ixn;

      s = (1.0f - k) * s;                // p = (I-K) p  (scalar*I)

      if (lane == 0) {
        const size_t r = (size_t)(c * 32 + i + 1) * 13;
        out[r + 0] = q0; out[r + 1] = q1; out[r + 2] = q2; out[r + 3] = q3;
        out[r + 4] = s;  out[r + 8] = s;  out[r + 12] = s;
      }
    }
  }
}

extern "C" void kernel_launch(void* const* d_in, const int* in_sizes, int n_in,
                              void* d_out, int out_size, void* d_ws, size_t ws_size,
                              hipStream_t stream) {
  (void)in_sizes; (void)n_in; (void)out_size; (void)d_ws; (void)ws_size;
  const float* x    = (const float*)d_in[0];   // (1, 32768, 6)
  const float* pred = (const float*)d_in[1];   // (1, 32768, 7)
  float* out = (float*)d_out;                  // (1, 32768, 13)

  ekf_fill_offdiag<<<256, 256, 0, stream>>>(out);
  ekf_scan<<<1, 32, 0, stream>>>(x, pred, out);
}